// MultiResHashGridMLP_33706903339713
// MI455X (gfx1250) — compile-verified
//
#include <hip/hip_runtime.h>

// Multi-resolution hash grid encoding (Instant-NGP style) — CDNA5 / gfx1250.
//
// The reference's xf = (xs - xs) == 0 collapses the trilinear weights to
// {1,0,0,0,0,0,0,0}: only the floor corner contributes, exactly. So:
//   out[n, 2l:2l+2] = tables[l][ hash(floor(x*res_l)) % hsize_l ]
// Pure gather kernel. Roofline plan for MI455X:
//   - 64 MB of tables fit in the 192 MB L2  -> gather with th:TH_LOAD_HT
//     (high-priority temporal) so the replacement policy pins them.
//   - x (24 MB) read once                  -> TH_LOAD_NT (no L2 pollution)
//   - out (256 MB) written once            -> TH_STORE_NT b128 full-line writes
//   - All 16 gathers issued as ONE hardware clause (s_clause 0xf) in SADDR
//     form: uniform SGPR-pair base + 32-bit byte offset (l<<22)+(h<<3),
//     drained by a single s_wait_loadcnt 0x0.

typedef float v2f __attribute__((ext_vector_type(2)));
typedef float v4f __attribute__((ext_vector_type(4)));

#define NLVL 16

// Per-level resolution: floor(16 * 2^(l/3))  (beta = 2^(1/3) exactly).
__device__ constexpr float kRes[NLVL] = {
    16.f, 20.f, 25.f, 32.f, 40.f, 50.f, 64.f, 80.f,
    101.f, 128.f, 161.f, 203.f, 256.f, 322.f, 406.f, 512.f};

// Per-level hashmap size: min(res^3, 2^19). Compile-time -> modulo folds to
// magic-multiply (or AND for pow2 levels 0,3,6,8..15).
__device__ constexpr unsigned kHSize[NLVL] = {
    4096u,   8000u,   15625u,  32768u,  64000u,  125000u, 262144u, 512000u,
    524288u, 524288u, 524288u, 524288u, 524288u, 524288u, 524288u, 524288u};

#define PRIME1 2654435761u
#define PRIME2 805459861u

__global__ __launch_bounds__(256) void
hashgrid_gather_kernel(const float* __restrict__ x,
                       const float* __restrict__ tables,
                       float* __restrict__ out, int n) {
    int i = blockIdx.x * blockDim.x + threadIdx.x;
    if (i >= n) return;

    // Streaming coord read (compiler merges to global_load_b96 th:TH_LOAD_NT).
    const float* xp = x + 3u * (unsigned)i;
    float px = __builtin_nontemporal_load(xp + 0);
    float py = __builtin_nontemporal_load(xp + 1);
    float pz = __builtin_nontemporal_load(xp + 2);

    // Phase 1: all hash math up front -> 16 byte-offsets into the table window.
    unsigned off[NLVL];
#pragma unroll
    for (int l = 0; l < NLVL; ++l) {
        // trunc-toward-zero matches .astype(int32) for x in [0,1)
        unsigned xi = (unsigned)(int)(px * kRes[l]);
        unsigned yi = (unsigned)(int)(py * kRes[l]);
        unsigned zi = (unsigned)(int)(pz * kRes[l]);
        unsigned h  = xi ^ (yi * PRIME1) ^ (zi * PRIME2);
        h = h % kHSize[l];
        // level slab = l * 2^19 entries * 8B = l << 22 ; entry = h * 8B
        off[l] = ((unsigned)l << 22) + (h << 3);
    }

    // Phase 2: one 16-load hardware clause of SADDR-form gathers with
    // high-priority-temporal hint (keeps the 64 MB table set L2-resident).
    // Single asm block guarantees no foreign instruction lands inside the
    // clause. Compiler waitcnt logic never sees these loads, so LOADcnt is
    // drained explicitly below before the results are consumed.
    v2f feat[NLVL];
    asm volatile(
        "s_clause 0xf\n\t"
        "global_load_b64 %0,  %16, %32 th:TH_LOAD_HT\n\t"
        "global_load_b64 %1,  %17, %32 th:TH_LOAD_HT\n\t"
        "global_load_b64 %2,  %18, %32 th:TH_LOAD_HT\n\t"
        "global_load_b64 %3,  %19, %32 th:TH_LOAD_HT\n\t"
        "global_load_b64 %4,  %20, %32 th:TH_LOAD_HT\n\t"
        "global_load_b64 %5,  %21, %32 th:TH_LOAD_HT\n\t"
        "global_load_b64 %6,  %22, %32 th:TH_LOAD_HT\n\t"
        "global_load_b64 %7,  %23, %32 th:TH_LOAD_HT\n\t"
        "global_load_b64 %8,  %24, %32 th:TH_LOAD_HT\n\t"
        "global_load_b64 %9,  %25, %32 th:TH_LOAD_HT\n\t"
        "global_load_b64 %10, %26, %32 th:TH_LOAD_HT\n\t"
        "global_load_b64 %11, %27, %32 th:TH_LOAD_HT\n\t"
        "global_load_b64 %12, %28, %32 th:TH_LOAD_HT\n\t"
        "global_load_b64 %13, %29, %32 th:TH_LOAD_HT\n\t"
        "global_load_b64 %14, %30, %32 th:TH_LOAD_HT\n\t"
        "global_load_b64 %15, %31, %32 th:TH_LOAD_HT"
        : "=v"(feat[0]),  "=v"(feat[1]),  "=v"(feat[2]),  "=v"(feat[3]),
          "=v"(feat[4]),  "=v"(feat[5]),  "=v"(feat[6]),  "=v"(feat[7]),
          "=v"(feat[8]),  "=v"(feat[9]),  "=v"(feat[10]), "=v"(feat[11]),
          "=v"(feat[12]), "=v"(feat[13]), "=v"(feat[14]), "=v"(feat[15])
        : "v"(off[0]),  "v"(off[1]),  "v"(off[2]),  "v"(off[3]),
          "v"(off[4]),  "v"(off[5]),  "v"(off[6]),  "v"(off[7]),
          "v"(off[8]),  "v"(off[9]),  "v"(off[10]), "v"(off[11]),
          "v"(off[12]), "v"(off[13]), "v"(off[14]), "v"(off[15]),
          "s"(tables));

    // All 16 gathers outstanding; drain before consuming. "memory" clobber
    // prevents the NT stores below from hoisting above this wait.
    asm volatile("s_wait_loadcnt 0x0" ::: "memory");

    // 128 B contiguous output row per point: 8x b128 non-temporal stores
    // (each lane covers its full cachelines -> no partial-line waste).
    v4f* op = (v4f*)(out + (size_t)i * (NLVL * 2));
#pragma unroll
    for (int l = 0; l < NLVL; l += 2) {
        v4f v = {feat[l].x, feat[l].y, feat[l + 1].x, feat[l + 1].y};
        __builtin_nontemporal_store(v, op + (l >> 1));
    }
}

extern "C" void kernel_launch(void* const* d_in, const int* in_sizes, int n_in,
                              void* d_out, int out_size, void* d_ws, size_t ws_size,
                              hipStream_t stream) {
    const float* x      = (const float*)d_in[0];   // [N, 3] float32
    const float* tables = (const float*)d_in[1];   // [16, 2^19, 2] float32
    float* out          = (float*)d_out;           // [N, 32] float32

    int n = in_sizes[0] / 3;
    const int threads = 256;                       // 8 wave32 per block
    int blocks = (n + threads - 1) / threads;
    hipLaunchKernelGGL(hashgrid_gather_kernel, dim3(blocks), dim3(threads), 0,
                       stream, x, tables, out, n);
}